// DWT_3D_89816356094282
// MI455X (gfx1250) — compile-verified
//
#include <hip/hip_runtime.h>

typedef __attribute__((ext_vector_type(2))) float v2f;
typedef __attribute__((ext_vector_type(8))) float v8f;

#define S3 0.35355339059327373f  // (1/sqrt(2))^3

// Haar 3D DWT, stride-2 depthwise, via V_WMMA_F32_16X16X4_F32.
// x: [2,16,32,256,256] f32 ; out: [2,128,16,128,128] f32 (flat == reference's view)
__global__ __launch_bounds__(256) void haar_dwt3d_wmma(const float* __restrict__ x,
                                                       float* __restrict__ out) {
    const int lane = threadIdx.x & 31;
    const int wave = threadIdx.x >> 5;
    int g = blockIdx.x * 8 + wave;          // 65536 waves total
    const int h2 = g & 127; g >>= 7;        // H/2 = 128
    const int d2 = g & 15;  g >>= 4;        // D/2 = 16
    const int c  = g & 15;  g >>= 4;        // C   = 16
    const int b  = g;                       // B   = 2

    const int half = (lane < 16) ? 0 : 1;   // which half of the wave
    const int col  = lane & 15;

    // ---- A operand: Haar sign matrix chunk (16x4), documented f32 A layout:
    // lanes 0-15 rows M=0-15 (VGPR0=K0, VGPR1=K1); lanes 16-31 rows M=0-15 (VGPR0=K2, VGPR1=K3).
    // A[m][t] = s^3 * (-1)^popc((m&7) & t); rows 8-15 therefore duplicate filters 0-7,
    // so D lanes 16-31 of VGPR r carry filter r for blocks 0-15 (a usable duplicate).
    const int f  = col & 7;
    const int t0 = half ? 2 : 0;
    v2f a1, a2;
    a1.x = (__popc(f & (t0 + 0)) & 1) ? -S3 : S3;
    a1.y = (__popc(f & (t0 + 1)) & 1) ? -S3 : S3;
    a2.x = (__popc(f & (t0 + 4)) & 1) ? -S3 : S3;   // taps 4-7: d-offset 1
    a2.y = (__popc(f & (t0 + 5)) & 1) ? -S3 : S3;

    // ---- input strides (floats)
    const long sH = 256, sD = 65536, sC = 2097152, sB = 33554432;
    // B operand (4x16): VGPR0 = K0(lanes0-15)/K2(lanes16-31), VGPR1 = K1/K3.
    // A float2 load at (2h2+half, 2w2) yields (.x,.y) = (K0,K1) / (K2,K3) per half.
    const long inBase = (long)b * sB + (long)c * sC + (long)(2 * d2) * sD
                      + (long)(2 * h2 + half) * sH + (long)(2 * col);

    // ---- output strides (floats): out[b][o][d2][h2][w2], o = 8c + filter
    const long oD2 = 16384, oO = 262144, oB = 33554432;
    // Two tiles per iteration: lanes<16 write tile t (w2 = 16t+col), lanes>=16 write
    // tile t+1 (w2 = 16t+16+col).  Fold half*16 into the base now.
    float* const outPtr = out + (long)b * oB + (long)(8 * c) * oO
                        + (long)d2 * oD2 + (long)h2 * 128 + half * 16 + col;

    for (int t = 0; t < 8; t += 2) {
        const float* p = x + inBase + 32 * t;        // tile t, plane d = 2*d2
        v2f bA0 = __builtin_nontemporal_load((const v2f*)p);
        v2f bA1 = __builtin_nontemporal_load((const v2f*)(p + sD));
        v2f bB0 = __builtin_nontemporal_load((const v2f*)(p + 32));       // tile t+1
        v2f bB1 = __builtin_nontemporal_load((const v2f*)(p + 32 + sD));

        v8f accA = {};
        accA = __builtin_amdgcn_wmma_f32_16x16x4_f32(false, a1, false, bA0,
                                                     (short)0, accA, false, false);
        accA = __builtin_amdgcn_wmma_f32_16x16x4_f32(false, a2, false, bA1,
                                                     (short)0, accA, false, false);
        v8f accB = {};
        accB = __builtin_amdgcn_wmma_f32_16x16x4_f32(false, a1, false, bB0,
                                                     (short)0, accB, false, false);
        accB = __builtin_amdgcn_wmma_f32_16x16x4_f32(false, a2, false, bB1,
                                                     (short)0, accB, false, false);

        // Per filter r: one fully contiguous 128B NT store covering w2 in [16t, 16t+32).
        // Select between the two tiles' accumulators at the SAME constant element index;
        // the asm barrier keeps the optimizer from refolding it into a variable-index
        // vector extract (the round-2 miscompile pattern).
        #pragma unroll
        for (int r = 0; r < 8; ++r) {
            float lo = accA[r];          // tile t,   filter r (lanes 0-15 view)
            float hi = accB[r];          // tile t+1, filter r (lanes 16-31 duplicate rows)
            asm("" : "+v"(lo));
            const float val = half ? hi : lo;
            __builtin_nontemporal_store(val, outPtr + (long)r * oO + t * 16);
        }
    }
}

extern "C" void kernel_launch(void* const* d_in, const int* in_sizes, int n_in,
                              void* d_out, int out_size, void* d_ws, size_t ws_size,
                              hipStream_t stream) {
    const float* x = (const float*)d_in[0];
    // d_in[1] = the 8x1x2x2x2 Haar filters; fixed by construction, hardcoded as signs above.
    float* out = (float*)d_out;

    // 2*16*16*128 = 65536 waves, 8 waves (256 threads) per block
    dim3 grid(8192), block(256);
    haar_dwt3d_wmma<<<grid, block, 0, stream>>>(x, out);
}